// BurstNeuron_36558761624016
// MI455X (gfx1250) — compile-verified
//
#include <hip/hip_runtime.h>

typedef float v4f __attribute__((ext_vector_type(4)));
typedef int   v4i __attribute__((ext_vector_type(4)));

#define THREADS 256
#define CHUNKS  16   // chunks (of THREADS float4) per block -> block spans 4 rows of C=4096
#define DEPTH   4    // async pipeline depth; LDS = DEPTH * THREADS * 16B = 16 KB

#if defined(__AMDGCN__) && __has_builtin(__builtin_amdgcn_global_load_async_to_lds_b128) && __has_builtin(__builtin_amdgcn_s_wait_asynccnt)
#define HAVE_ASYNC 1
#define AS1 __attribute__((address_space(1)))
#define AS3 __attribute__((address_space(3)))
#endif

// spike = min(k1, T) * th  ==  th * #{ j in 1..T : mem0 > j*th },  mem0 = th/2 + x
__device__ __forceinline__ v4f burst4(v4f xv, v4f t, int T) {
    v4f r;
#pragma unroll
    for (int j = 0; j < 4; ++j) {
        float th  = t[j];
        float mem = __fmaf_rn(th, 0.5f, xv[j]);
        float cnt = 0.0f;
        if (T == 4) {                     // fast path for the reference's T
            cnt += (mem >        th) ? 1.0f : 0.0f;
            cnt += (mem > 2.0f * th) ? 1.0f : 0.0f;
            cnt += (mem > 3.0f * th) ? 1.0f : 0.0f;
            cnt += (mem > 4.0f * th) ? 1.0f : 0.0f;
        } else {
            for (int jj = 1; jj <= T; ++jj)
                cnt += (mem > (float)jj * th) ? 1.0f : 0.0f;
        }
        r[j] = cnt * th;
    }
    return r;
}

// Specialized streaming kernel: requires C == 4096 and n divisible by THREADS*CHUNKS*4.
// Each block handles a contiguous span of CHUNKS*THREADS float4 (= 4 full rows of C),
// so thread tid's per-channel threshold float4 for chunk c is tk[c & 3].
__global__ __launch_bounds__(THREADS) void burst_pipe(const float* __restrict__ xg,
                                                      const float* __restrict__ thg,
                                                      const int*   __restrict__ Tp,
                                                      float*       __restrict__ outg) {
    const int  tid = threadIdx.x;
    const int  T   = *Tp;
    const v4f* x4  = (const v4f*)xg;
    v4f*       o4  = (v4f*)outg;
    const v4f* th4 = (const v4f*)thg;

    // Preload the 4 threshold vectors this thread will ever need (stays in registers;
    // the 16KB threshold array itself stays hot in WGP$/L2 across all blocks).
    v4f tk[4];
#pragma unroll
    for (int k = 0; k < 4; ++k) tk[k] = th4[k * THREADS + tid];

    const long base = (long)blockIdx.x * (THREADS * CHUNKS);

#if HAVE_ASYNC
    __shared__ v4f stage[DEPTH][THREADS];

    // Prologue: fill the pipeline (per-wave ASYNCcnt; each wave touches only its own slots,
    // so no workgroup barriers are needed anywhere).
#pragma unroll
    for (int s = 0; s < DEPTH; ++s)
        __builtin_amdgcn_global_load_async_to_lds_b128(
            (AS1 v4i*)(x4 + base + (long)s * THREADS + tid),
            (AS3 v4i*)&stage[s][tid], 0, 0);

#pragma unroll
    for (int c = 0; c < CHUNKS; ++c) {
        if (c < CHUNKS - DEPTH)
            __builtin_amdgcn_s_wait_asynccnt(DEPTH - 1);   // oldest in-flight chunk done
        else if (c == CHUNKS - DEPTH)
            __builtin_amdgcn_s_wait_asynccnt(0);           // drain for the tail

        v4f xv = stage[c % DEPTH][tid];                    // ds_load_b128
        v4f r  = burst4(xv, tk[c & 3], T);
        __builtin_nontemporal_store(r, o4 + base + (long)c * THREADS + tid);

        // xv's compiler-inserted s_wait_dscnt happened before this point; keep the
        // slot-reusing async issue ordered after the LDS read.
        asm volatile("" ::: "memory");
        if (c + DEPTH < CHUNKS)
            __builtin_amdgcn_global_load_async_to_lds_b128(
                (AS1 v4i*)(x4 + base + (long)(c + DEPTH) * THREADS + tid),
                (AS3 v4i*)&stage[c % DEPTH][tid], 0, 0);
    }
#else
    // Direct-load streaming fallback: non-temporal B128 in/out.
#pragma unroll 4
    for (int c = 0; c < CHUNKS; ++c) {
        v4f xv = __builtin_nontemporal_load(x4 + base + (long)c * THREADS + tid);
        v4f r  = burst4(xv, tk[c & 3], T);
        __builtin_nontemporal_store(r, o4 + base + (long)c * THREADS + tid);
    }
#endif
}

// Generic vectorized fallback (any C % 4 == 0, n % 4 == 0).
__global__ __launch_bounds__(THREADS) void burst_elem4(const float* __restrict__ xg,
                                                       const float* __restrict__ thg,
                                                       const int*   __restrict__ Tp,
                                                       float*       __restrict__ outg,
                                                       long n4, int C4) {
    const int T = *Tp;
    long i      = (long)blockIdx.x * blockDim.x + threadIdx.x;
    long stride = (long)gridDim.x * blockDim.x;
    for (; i < n4; i += stride) {
        v4f xv = __builtin_nontemporal_load((const v4f*)xg + i);
        v4f t  = ((const v4f*)thg)[(int)(i % C4)];
        __builtin_nontemporal_store(burst4(xv, t, T), (v4f*)outg + i);
    }
}

// Scalar fallback for arbitrary shapes.
__global__ __launch_bounds__(THREADS) void burst_elem1(const float* __restrict__ xg,
                                                       const float* __restrict__ thg,
                                                       const int*   __restrict__ Tp,
                                                       float*       __restrict__ outg,
                                                       long n, int C) {
    const int T = *Tp;
    long i = (long)blockIdx.x * blockDim.x + threadIdx.x;
    if (i >= n) return;
    float th  = thg[(int)(i % C)];
    float mem = __fmaf_rn(th, 0.5f, xg[i]);
    float cnt = 0.0f;
    for (int jj = 1; jj <= T; ++jj) cnt += (mem > (float)jj * th) ? 1.0f : 0.0f;
    outg[i] = cnt * th;
}

extern "C" void kernel_launch(void* const* d_in, const int* in_sizes, int n_in,
                              void* d_out, int out_size, void* d_ws, size_t ws_size,
                              hipStream_t stream) {
    const float* x   = (const float*)d_in[0];
    const float* th  = (const float*)d_in[1];
    const int*   Tp  = (const int*)d_in[2];
    float*       out = (float*)d_out;

    const long n = (long)in_sizes[0];
    const int  C = in_sizes[1];

    const long span = 4L * THREADS * CHUNKS;  // floats per block in the fast path
    if (C == 4096 && n % span == 0) {
        int blocks = (int)(n / span);         // 4096 blocks for the reference shape
        burst_pipe<<<blocks, THREADS, 0, stream>>>(x, th, Tp, out);
    } else if (n % 4 == 0 && C % 4 == 0) {
        long n4     = n / 4;
        long want   = (n4 + THREADS - 1) / THREADS;
        int  blocks = (int)(want < 262144 ? want : 262144);
        if (blocks < 1) blocks = 1;
        burst_elem4<<<blocks, THREADS, 0, stream>>>(x, th, Tp, out, n4, C / 4);
    } else {
        long blocks = (n + THREADS - 1) / THREADS;
        if (blocks < 1) blocks = 1;
        burst_elem1<<<(int)blocks, THREADS, 0, stream>>>(x, th, Tp, out, n, C);
    }
}